// BeamSearchDecoder_18829136625908
// MI455X (gfx1250) — compile-verified
//
#include <hip/hip_runtime.h>
#include <hip/hip_bf16.h>
#include <math.h>

// ---------------------------------------------------------------------------
// GRU encoder/decoder + attention + beam search (BEAM=4, SOS=1)
// All dense math runs through V_WMMA_F32_16X16X4_F32 (exact fp32 GEMM on the
// CDNA5 matrix pipe). M dimension padded to 16 rows (zeros beyond the 1 or 4
// live rows). Sequential scan structure expressed as a stream of small kernel
// launches (graph-capture safe).
// ---------------------------------------------------------------------------

typedef __attribute__((ext_vector_type(2))) float v2f;
typedef __attribute__((ext_vector_type(8))) float v8f;

__global__ void k_zero(float* p, long n) {
  long i = blockIdx.x * (long)blockDim.x + threadIdx.x;
  long st = (long)gridDim.x * blockDim.x;
  for (; i < n; i += st) p[i] = 0.f;
}

// dst[b][n] = emb[toks[b]][n], b < B (dst is a 16xH padded buffer; rows >= B
// stay zero from init).
__global__ void k_embed(const int* __restrict__ toks, const float* __restrict__ emb,
                        float* __restrict__ dst, int B, int H) {
  int i = blockIdx.x * blockDim.x + threadIdx.x;
  if (i >= B * H) return;
  int b = i / H, n = i - b * H;
  dst[b * H + n] = emb[(long)toks[b] * H + n];
}

// Generic fp32 WMMA GEMM:  D[gm][gn] = act( sum_k A[gm][k] * W[gn][k] + bias[gn] )
// A: (mTiles*16) x K row-major (rows beyond live data must be zero)
// W: (nTiles*16) x K row-major (weight matrix, "B" operand = W^T)
// Store only gm < outRows, gn < outCols into D (row-major, leading dim ldd).
// Per ISA 16x16x4 layout: lanes 0-15 carry K = {0,1}, lanes 16-31 K = {2,3};
// A rows and W rows use identical per-lane addressing.
__global__ void k_wmma_gemm(const float* __restrict__ A, const float* __restrict__ W,
                            const float* __restrict__ bias, float* __restrict__ D,
                            int K, int mTiles, int nTiles, int outRows, int outCols,
                            int ldd, int act) {
  int wid = (blockIdx.x * blockDim.x + threadIdx.x) >> 5;
  if (wid >= mTiles * nTiles) return;       // wave-uniform early-out (EXEC stays full)
  int lane = threadIdx.x & 31;
  int tm = wid / nTiles;
  int tn = wid - tm * nTiles;
  int rsel = lane & 15;
  int koff = (lane >> 4) << 1;              // 0 for lanes 0-15, 2 for lanes 16-31
  const float* Ar = A + ((long)(tm * 16 + rsel)) * K + koff;
  const float* Wr = W + ((long)(tn * 16 + rsel)) * K + koff;
  v8f c = {};
  for (int k = 0; k < K; k += 4) {
    v2f a = *(const v2f*)(Ar + k);
    v2f b = *(const v2f*)(Wr + k);
    c = __builtin_amdgcn_wmma_f32_16x16x4_f32(false, a, false, b, (short)0, c,
                                              false, false);
  }
  int gn = tn * 16 + rsel;
  float bv = 0.f;
  if (bias && gn < outCols) bv = bias[gn];
  int mbase = tm * 16 + ((lane >> 4) << 3);  // VGPR v -> M = v + 8*(lane>=16)
#pragma unroll
  for (int v = 0; v < 8; ++v) {
    int gm = mbase + v;
    if (gm < outRows && gn < outCols) {
      float val = c[v] + bv;
      if (act == 1) val = tanhf(val);
      D[(long)gm * ldd + gn] = val;
    }
  }
}

// GRU gate fusion: h' = (1-z)*n + z*h  with r,z from sigmoid, n from tanh.
// gi/gh are B x 3H (rows packed). outH may alias h (element-wise in place).
// copyOut (valid for B==1) mirrors the new hidden into enc_out[t].
__global__ void k_gru_combine(const float* __restrict__ gi, const float* __restrict__ gh,
                              const float* __restrict__ h, float* __restrict__ outH,
                              float* __restrict__ copyOut, int B, int H) {
  int i = blockIdx.x * blockDim.x + threadIdx.x;
  if (i >= B * H) return;
  int b = i / H, n = i - b * H;
  const float* gib = gi + (long)b * 3 * H;
  const float* ghb = gh + (long)b * 3 * H;
  float ir = gib[n], iz = gib[H + n], inn = gib[2 * H + n];
  float hr = ghb[n], hz = ghb[H + n], hn = ghb[2 * H + n];
  float r = 1.f / (1.f + __expf(-(ir + hr)));
  float z = 1.f / (1.f + __expf(-(iz + hz)));
  float nn = tanhf(inn + r * hn);
  float hv = h[b * H + n];
  float hnew = (1.f - z) * nn + z * hv;
  outH[b * H + n] = hnew;
  if (copyOut) copyOut[n] = hnew;
}

// Tile encoder final hidden across the 4 beams; reset beam state.
__global__ void k_init_beams(const float* __restrict__ hPe, float* __restrict__ hPd,
                             float* __restrict__ cum, int* __restrict__ inputs, int H) {
  int t = blockIdx.x * blockDim.x + threadIdx.x;
  if (t < 4) { cum[t] = 0.f; inputs[t] = 1; /* SOS */ }
  int stride = gridDim.x * blockDim.x;
  for (int i = t; i < 4 * H; i += stride) {
    int n = i & (H - 1);
    hPd[i] = hPe[n];
  }
}

// Softmax over s for each beam b (attn layout [s][b], ldd = 4).
__global__ void k_softmax(float* __restrict__ attn, int S) {
  __shared__ float red[256];
  int b = blockIdx.x, t = threadIdx.x;
  float m = -INFINITY;
  for (int s = t; s < S; s += 256) m = fmaxf(m, attn[s * 4 + b]);
  red[t] = m; __syncthreads();
  for (int off = 128; off > 0; off >>= 1) {
    if (t < off) red[t] = fmaxf(red[t], red[t + off]);
    __syncthreads();
  }
  m = red[0]; __syncthreads();
  float sum = 0.f;
  for (int s = t; s < S; s += 256) {
    float e = __expf(attn[s * 4 + b] - m);
    attn[s * 4 + b] = e;
    sum += e;
  }
  red[t] = sum; __syncthreads();
  for (int off = 128; off > 0; off >>= 1) {
    if (t < off) red[t] += red[t + off];
    __syncthreads();
  }
  float inv = 1.f / red[0];
  for (int s = t; s < S; s += 256) attn[s * 4 + b] *= inv;
}

// ctx[b][n] = sum_s attn[s][b] * encOut[s][n]
__global__ void k_ctx(const float* __restrict__ attn, const float* __restrict__ encOut,
                      float* __restrict__ ctx, int S, int H) {
  int i = blockIdx.x * blockDim.x + threadIdx.x;
  if (i >= 4 * H) return;
  int b = i / H, n = i - b * H;
  float acc = 0.f;
  for (int s = 0; s < S; ++s) acc += attn[s * 4 + b] * encOut[(long)s * H + n];
  ctx[b * H + n] = acc;
}

// concatP rows 0..3 = [hnew[b] | ctx[b]]; rows 4..15 stay zero from init.
__global__ void k_concat(const float* __restrict__ hnew, const float* __restrict__ ctx,
                         float* __restrict__ concatP, int H) {
  int i = blockIdx.x * blockDim.x + threadIdx.x;
  if (i >= 4 * 2 * H) return;
  int b = i / (2 * H), n = i - b * 2 * H;
  concatP[b * 2 * H + n] = (n < H) ? hnew[b * H + n] : ctx[b * H + (n - H)];
}

// Per-beam log-softmax + top-4 (value, index). blockIdx.x = beam.
__global__ void k_top4(const float* __restrict__ logits, float* __restrict__ tp,
                       int* __restrict__ ti, int V) {
  __shared__ float rv[256];
  __shared__ int ri[256];
  __shared__ int found[4];
  int b = blockIdx.x, t = threadIdx.x;
  const float* Lrow = logits + (long)b * V;
  float m = -INFINITY;
  for (int i = t; i < V; i += 256) m = fmaxf(m, Lrow[i]);
  rv[t] = m; __syncthreads();
  for (int off = 128; off > 0; off >>= 1) {
    if (t < off) rv[t] = fmaxf(rv[t], rv[t + off]);
    __syncthreads();
  }
  m = rv[0]; __syncthreads();
  float s = 0.f;
  for (int i = t; i < V; i += 256) s += __expf(Lrow[i] - m);
  rv[t] = s; __syncthreads();
  for (int off = 128; off > 0; off >>= 1) {
    if (t < off) rv[t] += rv[t + off];
    __syncthreads();
  }
  float lse = m + __logf(rv[0]);
  for (int j = 0; j < 4; ++j) {
    __syncthreads();
    float bm = -INFINITY;
    int bidx = -1;
    for (int i = t; i < V; i += 256) {
      bool skip = false;
      for (int q = 0; q < j; ++q) if (found[q] == i) skip = true;
      float v = Lrow[i];
      if (!skip && v > bm) { bm = v; bidx = i; }
    }
    rv[t] = bm; ri[t] = bidx; __syncthreads();
    for (int off = 128; off > 0; off >>= 1) {
      if (t < off) {
        bool take = (rv[t + off] > rv[t]) ||
                    (rv[t + off] == rv[t] && ri[t + off] >= 0 &&
                     (ri[t] < 0 || ri[t + off] < ri[t]));
        if (take) { rv[t] = rv[t + off]; ri[t] = ri[t + off]; }
      }
      __syncthreads();
    }
    if (t == 0) {
      found[j] = ri[0];
      tp[b * 4 + j] = rv[0] - lse;
      ti[b * 4 + j] = ri[0];
    }
  }
}

// Beam combine + state gather (single block). Matches the reference exactly,
// including new_h = hnew[jj].
__global__ void k_select(const float* __restrict__ tp, const int* __restrict__ ti,
                         float* __restrict__ cum, float* __restrict__ hPd,
                         const float* __restrict__ hnewP, int* __restrict__ inputs,
                         const int* __restrict__ tokOld, int* __restrict__ tokNew,
                         const float* __restrict__ shOld, float* __restrict__ shNew,
                         int step, int L, int H) {
  __shared__ int sbi[4], sjj[4], sinp[4];
  __shared__ float sstep[4];
  int t = threadIdx.x;
  if (t == 0) {
    float cand[16];
    for (int b = 0; b < 4; ++b)
      for (int j = 0; j < 4; ++j) {
        bool valid = (step == 0) ? (b == 0) : true;
        cand[b * 4 + j] = valid ? (cum[b] + tp[b * 4 + j]) : -INFINITY;
      }
    bool used[16];
    for (int c = 0; c < 16; ++c) used[c] = false;
    float newCum[4];
    for (int k = 0; k < 4; ++k) {
      int best = -1; float bv = -INFINITY;
      for (int c = 0; c < 16; ++c)
        if (!used[c] && (best < 0 || cand[c] > bv)) { bv = cand[c]; best = c; }
      used[best] = true;
      sbi[k] = best / 4;
      sjj[k] = best - (best / 4) * 4;
      newCum[k] = bv;
      sstep[k] = tp[best];
      sinp[k] = ti[best];
    }
    for (int k = 0; k < 4; ++k) { cum[k] = newCum[k]; inputs[k] = sinp[k]; }
  }
  __syncthreads();
  for (int idx = t; idx < 4 * L; idx += blockDim.x) {
    int k = idx / L, pos = idx - k * L;
    int v = tokOld[sbi[k] * L + pos];
    float sv = shOld[sbi[k] * L + pos];
    if (pos == step) { v = sinp[k]; sv = sstep[k]; }
    tokNew[k * L + pos] = v;
    shNew[k * L + pos] = sv;
  }
  for (int idx = t; idx < 4 * H; idx += blockDim.x) {
    int k = idx / H, n = idx - k * H;
    hPd[k * H + n] = hnewP[sjj[k] * H + n];   // reference: new_h = hnew[jj]
  }
}

// best = argmax(cum); out[0..L-1] = tokens[best] (as float), out[L..2L-1] = shist[best]
__global__ void k_finalize(const float* __restrict__ cum, const int* __restrict__ tok,
                           const float* __restrict__ sh, float* __restrict__ out, int L) {
  __shared__ int bestS;
  int t = threadIdx.x;
  if (t == 0) {
    int best = 0; float bv = cum[0];
    for (int b = 1; b < 4; ++b) if (cum[b] > bv) { bv = cum[b]; best = b; }
    bestS = best;
  }
  __syncthreads();
  for (int i = t; i < L; i += blockDim.x) {
    out[i] = (float)tok[bestS * L + i];
    out[L + i] = sh[bestS * L + i];
  }
}

static void launch_gemm(const float* A, const float* W, const float* bias, float* D,
                        int K, int mTiles, int nTiles, int outRows, int outCols,
                        int ldd, int act, hipStream_t stream) {
  int waves = mTiles * nTiles;
  int threads = waves * 32;
  int block = 256;
  int grid = (threads + block - 1) / block;
  k_wmma_gemm<<<grid, block, 0, stream>>>(A, W, bias, D, K, mTiles, nTiles, outRows,
                                          outCols, ldd, act);
}

extern "C" void kernel_launch(void* const* d_in, const int* in_sizes, int n_in,
                              void* d_out, int out_size, void* d_ws, size_t ws_size,
                              hipStream_t stream) {
  const int H = 1024, V = 32000, L = 48;   // L == max_length from setup_inputs()
  const int H3 = 3 * H;
  const int S = in_sizes[0];               // input_length == input_seq count

  const int* seq = (const int*)d_in[0];
  const float* emb  = (const float*)d_in[3];
  const float* eWih = (const float*)d_in[4];
  const float* eWhh = (const float*)d_in[5];
  const float* eBih = (const float*)d_in[6];
  const float* eBhh = (const float*)d_in[7];
  const float* dWih = (const float*)d_in[8];
  const float* dWhh = (const float*)d_in[9];
  const float* dBih = (const float*)d_in[10];
  const float* dBhh = (const float*)d_in[11];
  const float* Wc   = (const float*)d_in[12];
  const float* Wout = (const float*)d_in[13];

  // ---- workspace layout (floats; ints overlaid) ----
  float* ws = (float*)d_ws;
  size_t off = 0;
  auto F = [&](size_t n) { float* p = ws + off; off += n; return p; };
  float* xPe     = F((size_t)16 * H);       // encoder x, padded 16 rows
  float* hPe     = F((size_t)16 * H);       // encoder h, padded
  float* xPd     = F((size_t)16 * H);       // decoder x, padded
  float* hPd     = F((size_t)16 * H);       // decoder h, padded
  float* hnewP   = F((size_t)16 * H);       // decoder hnew, padded
  float* concatP = F((size_t)16 * 2 * H);   // [hnew|ctx], padded
  float* encOut  = F((size_t)S * H);
  float* gi      = F((size_t)4 * H3);
  float* gh      = F((size_t)4 * H3);
  float* attn    = F((size_t)S * 4);        // [s][b]
  float* ctx     = F((size_t)4 * H);
  float* ccP     = F((size_t)16 * H);       // tanh(concat @ Wc^T), padded
  float* logits  = F((size_t)4 * V);
  float* tp      = F(16);
  int*   ti      = (int*)F(16);
  float* cum     = F(8);
  int*   inputs  = (int*)F(8);
  int*   T0      = (int*)F((size_t)4 * L);
  int*   T1      = (int*)F((size_t)4 * L);
  float* S0      = F((size_t)4 * L);
  float* S1      = F((size_t)4 * L);
  size_t total = off;

  // Deterministic init: zero everything we use (padded rows must be zero).
  k_zero<<<512, 256, 0, stream>>>(ws, (long)total);

  // ---------------- encoder scan ----------------
  for (int t = 0; t < S; ++t) {
    k_embed<<<(H + 255) / 256, 256, 0, stream>>>(seq + t, emb, xPe, 1, H);
    launch_gemm(xPe, eWih, eBih, gi, H, 1, H3 / 16, 1, H3, H3, 0, stream);
    launch_gemm(hPe, eWhh, eBhh, gh, H, 1, H3 / 16, 1, H3, H3, 0, stream);
    k_gru_combine<<<(H + 255) / 256, 256, 0, stream>>>(
        gi, gh, hPe, hPe, encOut + (size_t)t * H, 1, H);
  }
  k_init_beams<<<16, 256, 0, stream>>>(hPe, hPd, cum, inputs, H);

  // ---------------- decoder beam-search scan ----------------
  for (int s = 0; s < L; ++s) {
    k_embed<<<(4 * H + 255) / 256, 256, 0, stream>>>(inputs, emb, xPd, 4, H);
    launch_gemm(xPd, dWih, dBih, gi, H, 1, H3 / 16, 4, H3, H3, 0, stream);
    launch_gemm(hPd, dWhh, dBhh, gh, H, 1, H3 / 16, 4, H3, H3, 0, stream);
    k_gru_combine<<<(4 * H + 255) / 256, 256, 0, stream>>>(
        gi, gh, hPd, hnewP, nullptr, 4, H);
    // scores[s][b] = enc_out[s] . hnew[b]
    launch_gemm(encOut, hnewP, nullptr, attn, H, (S + 15) / 16, 1, S, 4, 4, 0, stream);
    k_softmax<<<4, 256, 0, stream>>>(attn, S);
    k_ctx<<<(4 * H + 255) / 256, 256, 0, stream>>>(attn, encOut, ctx, S, H);
    k_concat<<<(4 * 2 * H + 255) / 256, 256, 0, stream>>>(hnewP, ctx, concatP, H);
    // tanh(concat @ Wc^T): store all 16 rows so ccP rows 4-15 = tanh(0) = 0
    launch_gemm(concatP, Wc, nullptr, ccP, 2 * H, 1, H / 16, 16, H, H, 1, stream);
    // logits = ccP @ Wout^T  (the big one: 2000 N-tiles x 256 WMMAs)
    launch_gemm(ccP, Wout, nullptr, logits, H, 1, V / 16, 4, V, V, 0, stream);
    k_top4<<<4, 256, 0, stream>>>(logits, tp, ti, V);
    const int* tOld = (s & 1) ? T1 : T0;
    int* tNew = (s & 1) ? T0 : T1;
    const float* shOld = (s & 1) ? S1 : S0;
    float* shNew = (s & 1) ? S0 : S1;
    k_select<<<1, 256, 0, stream>>>(tp, ti, cum, hPd, hnewP, inputs, tOld, tNew,
                                    shOld, shNew, s, L, H);
  }
  // L even -> final state sits in T0/S0
  k_finalize<<<1, 64, 0, stream>>>(cum, (L & 1) ? T1 : T0, (L & 1) ? S1 : S0,
                                   (float*)d_out, L);
}